// StructuralLoss_1898375545541
// MI455X (gfx1250) — compile-verified
//
#include <hip/hip_runtime.h>
#include <math.h>

typedef float v2f __attribute__((ext_vector_type(2)));
typedef float v8f __attribute__((ext_vector_type(8)));

#define H_IMG 4096
#define W_IMG 4096
#define TILE 16
#define WAVES_PER_BLOCK 4
#define RS 24   // stride of 22x22 staging region
#define QS 20   // stride (and padded extent) of q / yv rows

__device__ __forceinline__ float ld_img(const float* __restrict__ img, int gr, int gc) {
    return (gr >= 0 && gr < H_IMG && gc >= 0 && gc < W_IMG) ? img[gr * W_IMG + gc] : 0.0f;
}

__global__ void xcorr_init_kernel(unsigned long long* acc) { *acc = 0ull; }

__global__ void xcorr_fin_kernel(const unsigned long long* __restrict__ acc,
                                 float* __restrict__ out) {
    double mean = (double)(*acc) * (1.0 / 4294967296.0) / ((double)H_IMG * (double)W_IMG);
    out[0] = (float)mean;
    out[1] = (float)mean;
}

// Fused local-xcorr loss. One wave = one 16x16 output tile.
// Width-4 box filters (offsets -1..+2) run on the matrix pipe as banded-matrix
// multiplies: V_WMMA_F32_16X16X4_F32 chains (K=4 == box width).
__global__ __launch_bounds__(WAVES_PER_BLOCK * 32)
void xcorr_kernel(const float* __restrict__ img1,
                  const float* __restrict__ img2,
                  unsigned long long* __restrict__ accum)
{
    // per-wave LDS slices
    __shared__ float s_img[WAVES_PER_BLOCK][2][22][RS];   // 22x22 halo, both images
    __shared__ float s_q[WAVES_PER_BLOCK][3][20][QS];     // q on 19x19, row 19 / col 19 zero-padded
    __shared__ float s_tmp[WAVES_PER_BLOCK][960];         // h (2*22*20=880) then yv (3*16*20, col 19 zeroed)

    const int tid  = threadIdx.x;
    const int wave = tid >> 5;
    const int lane = tid & 31;
    const int half = lane >> 4;   // 0: lanes 0-15, 1: lanes 16-31
    const int l16  = lane & 15;
    const int K0   = half * 2;    // f32 A/B operand K-slot base for this half-wave

    const int R0 = blockIdx.y * TILE;
    const int C0 = blockIdx.x * (TILE * WAVES_PER_BLOCK) + wave * TILE;

    float (*si)[22][RS] = s_img[wave];
    float (*sq)[20][QS] = s_q[wave];
    float* st = s_tmp[wave];

    // Band operands, shared by BOTH box stages (vertical A-side, horizontal B-side):
    // band[c] slot K holds M[l16][4c+K] = 1 iff l16 <= 4c+K <= l16+3.
    v2f band[5];
#pragma unroll
    for (int c = 0; c < 5; ++c) {
        int r0 = 4 * c + K0;
        int r1 = r0 + 1;
        band[c].x = (r0 >= l16 && r0 <= l16 + 3) ? 1.0f : 0.0f;
        band[c].y = (r1 >= l16 && r1 <= l16 + 3) ? 1.0f : 0.0f;
    }

    // ---- 1. stage 22x22 halo (rows/cols R0-2..R0+19) for both images ----
#pragma unroll
    for (int f = lane; f < 22 * 22; f += 32) {
        int r = f / 22, c = f - r * 22;
        si[0][r][c] = ld_img(img1, R0 - 2 + r, C0 - 2 + c);
        si[1][r][c] = ld_img(img2, R0 - 2 + r, C0 - 2 + c);
    }
    __syncthreads();

    // ---- 2. horizontal width-4 sums: h[img][a][hj] = sum_{b=hj..hj+3} s[a][b] ----
    for (int f = lane; f < 2 * 22 * 19; f += 32) {
        int im  = f / (22 * 19);
        int rem = f - im * (22 * 19);
        int a   = rem / 19, hj = rem - a * 19;
        const float* row = si[im][a];
        st[im * 440 + a * 20 + hj] = row[hj] + row[hj + 1] + row[hj + 2] + row[hj + 3];
    }

    // zero-pad q: row 19 (all cols) and col 19 (rows 0..18) of each channel
    if (lane < 20) {
        sq[0][19][lane] = 0.0f; sq[1][19][lane] = 0.0f; sq[2][19][lane] = 0.0f;
    }
    if (lane < 19) {
        sq[0][lane][19] = 0.0f; sq[1][lane][19] = 0.0f; sq[2][lane][19] = 0.0f;
    }
    __syncthreads();

    // ---- 3. mu, centered values, products on 19x19 (q index (qi,qj) <-> region (qi+1,qj+1)) ----
    for (int f = lane; f < 19 * 19; f += 32) {
        int qi = f / 19, qj = f - qi * 19;
        int rr = qi + 1, cc = qj + 1;
        float mu1 = (st[qi * 20 + qj] + st[(qi + 1) * 20 + qj] +
                     st[(qi + 2) * 20 + qj] + st[(qi + 3) * 20 + qj]) * 0.0625f;
        float mu2 = (st[440 + qi * 20 + qj] + st[440 + (qi + 1) * 20 + qj] +
                     st[440 + (qi + 2) * 20 + qj] + st[440 + (qi + 3) * 20 + qj]) * 0.0625f;
        int gr = R0 - 1 + qi, gc = C0 - 1 + qj;
        bool inb = (gr >= 0 && gr < H_IMG && gc >= 0 && gc < W_IMG);
        float c1 = inb ? (si[0][rr][cc] - mu1) : 0.0f;   // zero-padded centered image
        float c2 = inb ? (si[1][rr][cc] - mu2) : 0.0f;
        sq[0][qi][qj] = c1 * c1;
        sq[1][qi][qj] = c2 * c2;
        sq[2][qi][qj] = c1 * c2;
    }
    __syncthreads();   // h is dead; st is reused as yv below

    // zero-pad yv col 19 so stage-5 A loads are unconditional
    if (lane < 16) {
        st[0 * 320 + lane * 20 + 19] = 0.0f;
        st[1 * 320 + lane * 20 + 19] = 0.0f;
        st[2 * 320 + lane * 20 + 19] = 0.0f;
    }

    // ---- 4. vertical box via WMMA: yv[ch][i][j] = sum_{r=i..i+3} q[ch][r][j] ----
    // D(16x16) = sum_{c=0..4} band_c(16x4) x Qchunk_c(4x16); two 16-col chunks cover 19 cols.
#pragma unroll
    for (int ch = 0; ch < 3; ++ch) {
#pragma unroll
        for (int cc2 = 0; cc2 < 2; ++cc2) {
            v8f acc = {0.f, 0.f, 0.f, 0.f, 0.f, 0.f, 0.f, 0.f};
            int col  = cc2 * 16 + l16;
            int colc = (col < 19) ? col : 19;   // cols >=19 read the zeroed pad slot
#pragma unroll
            for (int c = 0; c < 5; ++c) {
                int r0 = 4 * c + K0;            // rows 0..18 (.x), 1..19 (.y); row 19 zeroed
                v2f b;
                b.x = sq[ch][r0][colc];
                b.y = sq[ch][r0 + 1][colc];
                acc = __builtin_amdgcn_wmma_f32_16x16x4_f32(
                    false, band[c], false, b, (short)0, acc, false, false);
            }
            if (col < 19) {
#pragma unroll
                for (int v = 0; v < 8; ++v)
                    st[ch * 320 + (v + 8 * half) * 20 + col] = acc[v];  // D row = v + 8*half
            }
        }
    }
    __syncthreads();

    // ---- 5. horizontal box via WMMA: sig[ch](16x16) = sum_{c} Ychunk_c(16x4) x band_c(4x16) ----
    v8f sig[3];
#pragma unroll
    for (int ch = 0; ch < 3; ++ch) {
        v8f acc = {0.f, 0.f, 0.f, 0.f, 0.f, 0.f, 0.f, 0.f};
#pragma unroll
        for (int c = 0; c < 5; ++c) {
            int c0 = 4 * c + K0;                // cols 0..18 (.x), 1..19 (.y); col 19 zeroed
            v2f a;
            a.x = st[ch * 320 + l16 * 20 + c0];       // A row M = l16
            a.y = st[ch * 320 + l16 * 20 + c0 + 1];
            acc = __builtin_amdgcn_wmma_f32_16x16x4_f32(
                false, a, false, band[c], (short)0, acc, false, false);
        }
        sig[ch] = acc;
    }

    // ---- 6. final pointwise math directly on the C/D layout; all 3 sigs share slots ----
    double xs = 0.0;
#pragma unroll
    for (int v = 0; v < 8; ++v) {
        float sii = fmaxf(sig[0][v], 1e-20f);
        float sjj = fmaxf(sig[1][v], 1e-20f);
        float L   = sig[2][v] / (sqrtf(sii * sjj) + 1e-20f);
        L = fmaxf(L, -1.0f);
        xs += (double)(1.0f - L);
    }

    // wave32 reduction, then one deterministic fixed-point atomic per wave
    for (int off = 16; off > 0; off >>= 1)
        xs += __shfl_xor(xs, off, 32);
    if (lane == 0) {
        xs = fmax(xs, 0.0);
        atomicAdd(accum, (unsigned long long)(xs * 4294967296.0 + 0.5));
    }
}

extern "C" void kernel_launch(void* const* d_in, const int* in_sizes, int n_in,
                              void* d_out, int out_size, void* d_ws, size_t ws_size,
                              hipStream_t stream) {
    const float* img1 = (const float*)d_in[0];
    const float* img2 = (const float*)d_in[1];
    unsigned long long* acc = (unsigned long long*)d_ws;
    float* out = (float*)d_out;

    xcorr_init_kernel<<<dim3(1), dim3(1), 0, stream>>>(acc);

    dim3 grid(W_IMG / (TILE * WAVES_PER_BLOCK), H_IMG / TILE);  // 64 x 256
    xcorr_kernel<<<grid, dim3(WAVES_PER_BLOCK * 32), 0, stream>>>(img1, img2, acc);

    xcorr_fin_kernel<<<dim3(1), dim3(1), 0, stream>>>(acc, out);
}